// GraphSAGE_5866925326494
// MI455X (gfx1250) — compile-verified
//
#include <hip/hip_runtime.h>

#define HID 256
#define EMB 128

typedef float v2f __attribute__((ext_vector_type(2)));
typedef float v8f __attribute__((ext_vector_type(8)));

// Non-returning fp32 atomic add -> global_atomic_add_f32 (L2-side RMW).
__device__ __forceinline__ void atomic_add_f32(float* p, float v) {
    unsafeAtomicAdd(p, v);
}

// ------------------------------------------------------------------
// x[i,:] = emb[node_ids[i],:]   (float4 vectorized gather)
// ------------------------------------------------------------------
__global__ void gather_x_kernel(const float* __restrict__ emb,
                                const int* __restrict__ nid,
                                float* __restrict__ x, int n) {
    int g = blockIdx.x * blockDim.x + threadIdx.x;   // one float4 per thread
    int total = n * (HID / 4);
    if (g >= total) return;
    int row = g / (HID / 4);
    int j   = g - row * (HID / 4);
    const float4* s = (const float4*)(emb + (size_t)nid[row] * HID);
    float4* d       = (float4*)(x + (size_t)row * HID);
    d[j] = s[j];
}

// ------------------------------------------------------------------
// cnt[dst[e]] += 1.0f
// ------------------------------------------------------------------
__global__ void count_kernel(const int* __restrict__ dst,
                             float* __restrict__ cnt, int E) {
    int e = blockIdx.x * blockDim.x + threadIdx.x;
    if (e < E) atomic_add_f32(&cnt[dst[e]], 1.0f);
}

// ------------------------------------------------------------------
// agg[dst[e],:] += w[e] * x[src[e],:]
// One wave (32 lanes) per edge, 8 floats per lane (2x float4 loads).
// x/agg are L2-resident (51 MB each vs 192 MB L2).
// ------------------------------------------------------------------
__global__ void scatter_kernel(const float* __restrict__ x,
                               const int* __restrict__ src,
                               const int* __restrict__ dst,
                               const float* __restrict__ w,
                               float* __restrict__ agg, int E) {
    long long g = (long long)blockIdx.x * blockDim.x + threadIdx.x;
    int e    = (int)(g >> 5);
    int lane = (int)(g & 31);
    if (e >= E) return;
    int s = src[e], d = dst[e];
    float wt = w[e];
    const float4* xs = (const float4*)(x + (size_t)s * HID);
    float4 v0 = xs[lane];
    float4 v1 = xs[lane + 32];
    float* ag = agg + (size_t)d * HID + lane * 4;
    atomic_add_f32(ag + 0,   v0.x * wt);
    atomic_add_f32(ag + 1,   v0.y * wt);
    atomic_add_f32(ag + 2,   v0.z * wt);
    atomic_add_f32(ag + 3,   v0.w * wt);
    atomic_add_f32(ag + 128, v1.x * wt);
    atomic_add_f32(ag + 129, v1.y * wt);
    atomic_add_f32(ag + 130, v1.z * wt);
    atomic_add_f32(ag + 131, v1.w * wt);
}

// ------------------------------------------------------------------
// out = act( (agg @ Wl) * inv  +  x @ Wr  +  bias ),  inv = 1/max(cnt,1)
// (normalization is per-row linear, so it is applied to the pass-1
//  accumulators once instead of scaling every A fragment in the loop)
//
// Each wave owns an (MT*16) x 16 block of C: the B fragment is loaded
// once per k-step and reused across MT WMMAs; A fragments are vector
// b64 loads feeding the WMMAs directly (no VALU in the hot loop).
//
// Fragment layouts per CDNA5 ISA 7.12.2:
//  A 16x4 f32 : lanes 0-15 -> M=lane,    v0=K+0, v1=K+1
//               lanes 16-31 -> M=lane-16, v0=K+2, v1=K+3
//  B 4x16 f32 : mirrored (row K striped across lanes in a VGPR)
//  C/D 16x16  : VGPR r: lanes 0-15 -> M=r, N=lane; lanes 16-31 -> M=r+8
// ------------------------------------------------------------------
template <int NC, bool RELU, int MT>
__global__ __launch_bounds__(256)
void sage_gemm_kernel(const float* __restrict__ agg,
                      const float* __restrict__ cnt,
                      const float* __restrict__ x,
                      const float* __restrict__ Wl,
                      const float* __restrict__ Wr,
                      const float* __restrict__ bias,
                      float* __restrict__ out, int nrows) {
    const int ntn = NC / 16;                         // N tiles
    int wid  = blockIdx.x * (blockDim.x >> 5) + (threadIdx.x >> 5);
    int lane = threadIdx.x & 31;
    int nMblk = (nrows + MT * 16 - 1) / (MT * 16);
    int bm = wid / ntn;                              // M block (MT tiles)
    int tn = wid - bm * ntn;
    if (bm >= nMblk) return;

    int laneM = lane & 15;
    int kOff  = (lane >> 4) << 1;                    // 0 or 2
    int col   = tn * 16 + laneM;

    const float* Ar[MT];
    const float* Xr[MT];
#pragma unroll
    for (int i = 0; i < MT; ++i) {
        int r = bm * MT * 16 + i * 16 + laneM;
        if (r >= nrows) r = nrows - 1;               // clamp (exact for N=50000)
        Ar[i] = agg + (size_t)r * HID;
        Xr[i] = x   + (size_t)r * HID;
    }

    v8f c[MT];
#pragma unroll
    for (int i = 0; i < MT; ++i)
        c[i] = (v8f){0.f, 0.f, 0.f, 0.f, 0.f, 0.f, 0.f, 0.f};

    // ---- pass 1: agg @ Wl  (raw fragments, loads feed WMMA directly) ----
#pragma unroll 4
    for (int k = 0; k < HID; k += 4) {
        v2f b;
        b.x = Wl[(size_t)(k + kOff)     * NC + col];
        b.y = Wl[(size_t)(k + kOff + 1) * NC + col];
#pragma unroll
        for (int i = 0; i < MT; ++i) {
            v2f a = *(const v2f*)(Ar[i] + k + kOff);
            c[i] = __builtin_amdgcn_wmma_f32_16x16x4_f32(false, a, false, b,
                                                         (short)0, c[i], false, false);
        }
    }

    // ---- normalize pass-1 accumulators: c[i][r] *= 1/max(cnt[row],1) ----
    // C layout: VGPR r holds row (tile_base + r + 8*(lane>=16)).
#pragma unroll
    for (int i = 0; i < MT; ++i) {
        int mbase = bm * MT * 16 + i * 16 + ((lane >> 4) << 3);
#pragma unroll
        for (int r = 0; r < 8; ++r) {
            int m = mbase + r;
            if (m >= nrows) m = nrows - 1;
            c[i][r] *= 1.0f / fmaxf(cnt[m], 1.0f);
        }
    }

    // ---- pass 2: x @ Wr ----
#pragma unroll 4
    for (int k = 0; k < HID; k += 4) {
        v2f b;
        b.x = Wr[(size_t)(k + kOff)     * NC + col];
        b.y = Wr[(size_t)(k + kOff + 1) * NC + col];
#pragma unroll
        for (int i = 0; i < MT; ++i) {
            v2f a = *(const v2f*)(Xr[i] + k + kOff);
            c[i] = __builtin_amdgcn_wmma_f32_16x16x4_f32(false, a, false, b,
                                                         (short)0, c[i], false, false);
        }
    }

    float bv = bias[col];
#pragma unroll
    for (int i = 0; i < MT; ++i) {
        int mbase = bm * MT * 16 + i * 16 + ((lane >> 4) << 3);  // lanes 16-31: M += 8
#pragma unroll
        for (int r = 0; r < 8; ++r) {
            int m = mbase + r;
            if (m >= nrows) continue;
            float v = c[i][r] + bv;
            if (RELU) v = fmaxf(v, 0.0f);
            out[(size_t)m * NC + col] = v;
        }
    }
}

// ------------------------------------------------------------------
extern "C" void kernel_launch(void* const* d_in, const int* in_sizes, int n_in,
                              void* d_out, int out_size, void* d_ws, size_t ws_size,
                              hipStream_t stream) {
    const int*   node_ids = (const int*)  d_in[0];
    const int*   edge_idx = (const int*)  d_in[1];
    const float* ew       = (const float*)d_in[2];
    const float* emb      = (const float*)d_in[3];
    const float* W1l      = (const float*)d_in[4];
    const float* W1r      = (const float*)d_in[5];
    const float* b1       = (const float*)d_in[6];
    const float* W2l      = (const float*)d_in[7];
    const float* W2r      = (const float*)d_in[8];
    const float* b2       = (const float*)d_in[9];

    const int N = in_sizes[0];
    const int E = in_sizes[1] / 2;
    const int* src = edge_idx;
    const int* dst = edge_idx + E;

    float* ws  = (float*)d_ws;
    size_t NF  = (size_t)N * HID;
    float* x   = ws;                 // N x 256
    float* agg = ws + NF;            // N x 256 (reused across layers)
    float* h1  = ws + 2 * NF;        // N x 256
    float* cnt = ws + 3 * NF;        // N

    constexpr int MT = 5;            // 80 rows per wave; 50000/80 = 625 exactly

    hipMemsetAsync(agg, 0, NF * sizeof(float), stream);
    hipMemsetAsync(cnt, 0, (size_t)N * sizeof(float), stream);

    {   // x = emb[node_ids]
        int tot = N * (HID / 4);
        gather_x_kernel<<<(tot + 255) / 256, 256, 0, stream>>>(emb, node_ids, x, N);
    }
    count_kernel<<<(E + 255) / 256, 256, 0, stream>>>(dst, cnt, E);

    {   // layer-1 aggregation
        long long tot = (long long)E * 32;
        scatter_kernel<<<(unsigned)((tot + 255) / 256), 256, 0, stream>>>(x, src, dst, ew, agg, E);
    }
    {   // h1 = relu((agg @ W1l)*inv + x @ W1r + b1)
        int nMblk  = (N + MT * 16 - 1) / (MT * 16);
        int waves  = nMblk * (HID / 16);
        int blocks = (waves + 7) / 8;                // 8 waves / block
        sage_gemm_kernel<HID, true, MT><<<blocks, 256, 0, stream>>>(agg, cnt, x, W1l, W1r, b1, h1, N);
    }

    hipMemsetAsync(agg, 0, NF * sizeof(float), stream);
    {   // layer-2 aggregation on h1
        long long tot = (long long)E * 32;
        scatter_kernel<<<(unsigned)((tot + 255) / 256), 256, 0, stream>>>(h1, src, dst, ew, agg, E);
    }
    {   // out = (agg @ W2l)*inv + h1 @ W2r + b2
        int nMblk  = (N + MT * 16 - 1) / (MT * 16);
        int waves  = nMblk * (EMB / 16);
        int blocks = (waves + 7) / 8;
        sage_gemm_kernel<EMB, false, MT><<<blocks, 256, 0, stream>>>(agg, cnt, h1, W2l, W2r, b2,
                                                                    (float*)d_out, N);
    }
}